// GraphGNNModel_13786845020550
// MI455X (gfx1250) — compile-verified
//
#include <hip/hip_runtime.h>

#define N_NODES  50000   // == 3125 * 16, so 16-row tiles cover N exactly
#define N_EDGES  800000
#define CIN0     64
#define CH       128
#define COUT     2
#define NGRAPH   512
#define NEG_SLOPE 0.01f

typedef __attribute__((ext_vector_type(2))) float v2f;
typedef __attribute__((ext_vector_type(8))) float v8f;

// ---------------- utility kernels ----------------

__global__ void zero_f32(float* __restrict__ p, int n) {
  int i = blockIdx.x * blockDim.x + threadIdx.x;
  if (i < n) p[i] = 0.0f;
}

__global__ void edge_count_kernel(const int* __restrict__ dst, float* __restrict__ cnt) {
  int e = blockIdx.x * blockDim.x + threadIdx.x;
  if (e < N_EDGES) atomicAdd(&cnt[dst[e]], 1.0f);
}

__global__ void graph_count_kernel(const int* __restrict__ batch, float* __restrict__ gcnt) {
  int i = blockIdx.x * blockDim.x + threadIdx.x;
  if (i < N_NODES) atomicAdd(&gcnt[batch[i]], 1.0f);
}

// Pack weight matrix W (2*pairs x CH, row-major) into K-pair float2 layout:
// pk[p*CH + n] = { W[2p][n], W[2p+1][n] }  -> one b64 load per WMMA B-fragment.
__global__ void pack_w_kernel(const float* __restrict__ W, v2f* __restrict__ pk, int pairs) {
  int t = blockIdx.x * blockDim.x + threadIdx.x;
  if (t >= pairs * CH) return;
  int p = t / CH, n = t - p * CH;
  v2f v;
  v.x = W[(2 * p) * CH + n];
  v.y = W[(2 * p + 1) * CH + n];
  pk[t] = v;
}

// msum[dst[e]][c..c+3] += h[src[e]][c..c+3]
__global__ void scatter_add_kernel(const float* __restrict__ h,
                                   const int* __restrict__ src,
                                   const int* __restrict__ dst,
                                   float* __restrict__ msum, int C) {
  int t  = blockIdx.x * blockDim.x + threadIdx.x;
  int qn = C >> 2;
  int e  = t / qn;
  int q  = (t - e * qn) << 2;
  if (e >= N_EDGES) return;
  const float4 v = *reinterpret_cast<const float4*>(h + (size_t)src[e] * C + q);
  float* o = msum + (size_t)dst[e] * C + q;
  atomicAdd(o + 0, v.x);
  atomicAdd(o + 1, v.y);
  atomicAdd(o + 2, v.z);
  atomicAdd(o + 3, v.w);
}

// ---------------- fused SAGEConv GEMM (WMMA f32 16x16x4) ----------------
// out[row,:CH] = act( agg(row,:) @ Wl + h(row,:) @ Wr + b ),  agg = msum/cnt (0 if cnt==0)
// Block: 256 threads = 8 waves; tile = 16 nodes x 128 cols; wave w -> cols [16w,16w+16).
// A-frag (ISA 7.12.2, 32-bit A 16x4): lanes 0-15 hold {K=k,k+1} of row m=lane;
// lanes 16-31 hold {K=k+2,k+3}. B-frag: col n=lane%16, same K split.
// ACT: 0=none 1=relu 2=leaky
template <int CIN, int ACT>
__global__ __launch_bounds__(256)
void sage_gemm_kernel(const float* __restrict__ msum, const float* __restrict__ cnt,
                      const float* __restrict__ h,
                      const v2f* __restrict__ Wlp, const v2f* __restrict__ Wrp,
                      const float* __restrict__ bias, float* __restrict__ out) {
  constexpr int K2  = 2 * CIN;
  constexpr int STR = 2 * CH + 2;   // +2 pad: 8B-aligned pair reads, conflict-free
  __shared__ float As[16 * STR];

  const int row0 = blockIdx.x * 16;
  const int tid  = threadIdx.x;
  const int lane = tid & 31;
  const int wave = tid >> 5;

  // Stage A = [normalized-agg | h] for 16 rows (no bounds checks: N % 16 == 0).
#pragma unroll
  for (int it = 0; it < (16 * K2) / 256; ++it) {
    int idx = tid + it * 256;
    int r = idx / K2;
    int c = idx - r * K2;
    int node = row0 + r;
    float v;
    if (c < CIN) {
      float ct = cnt[node];
      v = (ct > 0.0f) ? msum[(size_t)node * CIN + c] / ct : 0.0f;
    } else {
      v = h[(size_t)node * CIN + (c - CIN)];
    }
    As[r * STR + c] = v;
  }
  __syncthreads();

  const int n0    = wave << 4;
  const int nn    = lane & 15;
  const int m     = lane & 15;
  const int khalf = (lane < 16) ? 0 : 2;  // lane-half K split
  const int phalf = khalf >> 1;           // pair index split

  const float* aptr  = As + m * STR + khalf;             // even offset -> 8B aligned
  const v2f*   blptr = Wlp + (size_t)phalf * CH + n0 + nn;
  const v2f*   brptr = Wrp + (size_t)phalf * CH + n0 + nn;

  v8f acc = {0.f, 0.f, 0.f, 0.f, 0.f, 0.f, 0.f, 0.f};

  // First half: agg @ Wl
#pragma unroll
  for (int k = 0; k < CIN; k += 4) {
    v2f a = *reinterpret_cast<const v2f*>(aptr + k);
    v2f b = blptr[(k >> 1) * CH];
    acc = __builtin_amdgcn_wmma_f32_16x16x4_f32(false, a, false, b,
                                                (short)0, acc, false, false);
  }
  // Second half: h @ Wr
#pragma unroll
  for (int k = 0; k < CIN; k += 4) {
    v2f a = *reinterpret_cast<const v2f*>(aptr + CIN + k);
    v2f b = brptr[(k >> 1) * CH];
    acc = __builtin_amdgcn_wmma_f32_16x16x4_f32(false, a, false, b,
                                                (short)0, acc, false, false);
  }

  // Epilogue: bias + activation + store (branch-free per instantiation).
  const float bcol  = bias[n0 + nn];
  const int   mbase = (lane < 16) ? 0 : 8;
  float* op = out + (size_t)(row0 + mbase) * CH + n0 + nn;
#pragma unroll
  for (int j = 0; j < 8; ++j) {
    float v = acc[j] + bcol;
    if (ACT == 1) v = (v > 0.0f) ? v : 0.0f;
    if (ACT == 2) v = (v > 0.0f) ? v : NEG_SLOPE * v;
    op[(size_t)j * CH] = v;
  }
}

// ---------------- pooling + head ----------------

__global__ void pool_scatter_kernel(const float* __restrict__ h,
                                    const int* __restrict__ batch,
                                    float* __restrict__ gsum) {
  int t = blockIdx.x * blockDim.x + threadIdx.x;  // N*32 threads, 4 cols each
  int node = t >> 5;
  int q = (t & 31) << 2;
  if (node >= N_NODES) return;
  int g = batch[node];
  const float4 v = *reinterpret_cast<const float4*>(h + (size_t)node * CH + q);
  float* o = gsum + (size_t)g * CH + q;
  atomicAdd(o + 0, v.x);
  atomicAdd(o + 1, v.y);
  atomicAdd(o + 2, v.z);
  atomicAdd(o + 3, v.w);
}

__global__ void head_kernel(const float* __restrict__ gsum, const float* __restrict__ gcnt,
                            const float* __restrict__ hW, const float* __restrict__ hb,
                            float* __restrict__ out) {
  int t = blockIdx.x * blockDim.x + threadIdx.x;
  if (t >= NGRAPH * COUT) return;
  int g = t / COUT, co = t - g * COUT;
  float inv = 1.0f / fmaxf(gcnt[g], 1.0f);
  float s = 0.0f;
#pragma unroll 8
  for (int k = 0; k < CH; ++k)
    s += gsum[(size_t)g * CH + k] * inv * hW[k * COUT + co];
  out[t] = s + hb[co];
}

// ---------------- host side ----------------

static inline int cdiv(long long a, int b) { return (int)((a + b - 1) / b); }

extern "C" void kernel_launch(void* const* d_in, const int* in_sizes, int n_in,
                              void* d_out, int out_size, void* d_ws, size_t ws_size,
                              hipStream_t stream) {
  (void)in_sizes; (void)n_in; (void)out_size; (void)ws_size;
  const float* x     = (const float*)d_in[0];
  const int*   eidx  = (const int*)  d_in[1];
  const int*   batch = (const int*)  d_in[2];
  const float* Wl0   = (const float*)d_in[3];
  const float* Wr0   = (const float*)d_in[4];
  const float* b0    = (const float*)d_in[5];
  const float* Wl    = (const float*)d_in[6];   // (11, CH, CH)
  const float* Wr    = (const float*)d_in[7];
  const float* bb    = (const float*)d_in[8];   // (11, CH)
  const float* hW    = (const float*)d_in[9];
  const float* hb    = (const float*)d_in[10];
  float* out = (float*)d_out;

  const int* src = eidx;            // edge_index[0]
  const int* dst = eidx + N_EDGES;  // edge_index[1]

  // Workspace layout (floats)
  float* ws   = (float*)d_ws;
  float* msum = ws;                                   // N*CH
  float* hA   = msum + (size_t)N_NODES * CH;          // N*CH
  float* hB   = hA   + (size_t)N_NODES * CH;          // N*CH
  float* cnt  = hB   + (size_t)N_NODES * CH;          // N
  float* gsum = cnt  + N_NODES;                       // G*CH
  float* gcnt = gsum + (size_t)NGRAPH * CH;           // G
  float* pkf  = gcnt + NGRAPH;                        // packed weights
  // layer 0: 2 matrices of 32 pairs x CH; layers 1..11: 2 matrices of 64 pairs x CH
  v2f* pk0l = (v2f*)pkf;                              // 32*CH v2f
  v2f* pk0r = pk0l + 32 * CH;
  v2f* pkLR = pk0r + 32 * CH;                         // 11 layers x 2 x (64*CH)

  // Zero counters + pooling buffers (contiguous region starting at cnt).
  {
    int n = N_NODES + NGRAPH * CH + NGRAPH;
    zero_f32<<<cdiv(n, 256), 256, 0, stream>>>(cnt, n);
  }
  edge_count_kernel<<<cdiv(N_EDGES, 256), 256, 0, stream>>>(dst, cnt);
  graph_count_kernel<<<cdiv(N_NODES, 256), 256, 0, stream>>>(batch, gcnt);

  // Pack weights into K-pair layout.
  pack_w_kernel<<<cdiv(32 * CH, 256), 256, 0, stream>>>(Wl0, pk0l, 32);
  pack_w_kernel<<<cdiv(32 * CH, 256), 256, 0, stream>>>(Wr0, pk0r, 32);
  for (int i = 0; i < 11; ++i) {
    pack_w_kernel<<<cdiv(64 * CH, 256), 256, 0, stream>>>(
        Wl + (size_t)i * CH * CH, pkLR + (size_t)(2 * i) * 64 * CH, 64);
    pack_w_kernel<<<cdiv(64 * CH, 256), 256, 0, stream>>>(
        Wr + (size_t)i * CH * CH, pkLR + (size_t)(2 * i + 1) * 64 * CH, 64);
  }

  const int gemm_blocks = N_NODES / 16;  // exact: 3125

  // ---- Layer 0 (Cin = 64, relu) ----
  {
    int n = N_NODES * CIN0;
    zero_f32<<<cdiv(n, 256), 256, 0, stream>>>(msum, n);
    long long th = (long long)N_EDGES * (CIN0 / 4);
    scatter_add_kernel<<<cdiv(th, 256), 256, 0, stream>>>(x, src, dst, msum, CIN0);
    sage_gemm_kernel<CIN0, 1><<<gemm_blocks, 256, 0, stream>>>(msum, cnt, x,
                                                               pk0l, pk0r, b0, hA);
  }

  // ---- Layers 1..11 (Cin = 128) ----
  // Activation per conv index 1..11: relu,relu,leaky,relu,relu,relu,leaky,relu,relu,relu,none
  const int acts[11] = {1, 1, 2, 1, 1, 1, 2, 1, 1, 1, 0};
  float* cur = hA;
  float* nxt = hB;
  for (int i = 0; i < 11; ++i) {
    int n = N_NODES * CH;
    zero_f32<<<cdiv(n, 256), 256, 0, stream>>>(msum, n);
    long long th = (long long)N_EDGES * (CH / 4);
    scatter_add_kernel<<<cdiv(th, 256), 256, 0, stream>>>(cur, src, dst, msum, CH);
    v2f* wl = pkLR + (size_t)(2 * i) * 64 * CH;
    v2f* wr = pkLR + (size_t)(2 * i + 1) * 64 * CH;
    const float* bi = bb + (size_t)i * CH;
    switch (acts[i]) {
      case 0:
        sage_gemm_kernel<CH, 0><<<gemm_blocks, 256, 0, stream>>>(msum, cnt, cur, wl, wr, bi, nxt);
        break;
      case 1:
        sage_gemm_kernel<CH, 1><<<gemm_blocks, 256, 0, stream>>>(msum, cnt, cur, wl, wr, bi, nxt);
        break;
      default:
        sage_gemm_kernel<CH, 2><<<gemm_blocks, 256, 0, stream>>>(msum, cnt, cur, wl, wr, bi, nxt);
        break;
    }
    float* t = cur; cur = nxt; nxt = t;
  }

  // ---- global mean pool + head ----
  {
    long long th = (long long)N_NODES * 32;
    pool_scatter_kernel<<<cdiv(th, 256), 256, 0, stream>>>(cur, batch, gsum);
    head_kernel<<<cdiv(NGRAPH * COUT, 256), 256, 0, stream>>>(gsum, gcnt, hW, hb, out);
  }
}